// GENEO_3848290697800
// MI455X (gfx1250) — compile-verified
//
#include <hip/hip_runtime.h>
#include <cmath>

// GENEO reduced pipeline for MI455X (gfx1250, wave32):
//   1) geneo1 slice: F[8][28][1536] thresholded (thr has closed form: window at (0,0) is all padding)
//   2) Wmat build:   Wmat[1536][224] from bilinear corner weights (rows clip to 0..27)
//   3) GEMM+max:     Tk = Wmat x F via V_WMMA_F32_16X16X4_F32, fused per-tile max
//   4) final max reduce -> scalar

typedef __attribute__((ext_vector_type(2))) float v2f;
typedef __attribute__((ext_vector_type(8))) float v8f;

#define IMG_W 1536
#define IMG_H 1536
#define KPAT 8
#define NROWS 28            // CLIP_MAX + 1
#define KDIM (KPAT * NROWS) // 224
#define NTILE 96            // 1536 / 16

// ---------------- Kernel 1: geneo1 slice (rows 0..27 only) ----------------
__global__ void geneo1_slice_kernel(const float* __restrict__ x,
                                    const float* __restrict__ patterns,
                                    float* __restrict__ F) {
  int idx = blockIdx.x * blockDim.x + threadIdx.x;       // over KPAT*NROWS*IMG_W
  int k   = idx / (NROWS * IMG_W);
  int rem = idx - k * (NROWS * IMG_W);
  int i   = rem / IMG_W;
  int j   = rem - i * IMG_W;

  const float* pat = patterns + k * 49;
  float acc = 0.f;   // sum |p - win| for this (i,j)
  float psum = 0.f;  // sum |p| -> threshold (window at (0,0) is all zero padding)
#pragma unroll
  for (int a = 0; a < 7; ++a) {
    int row = i + a - 7;   // always < 27, may be negative (zero pad)
#pragma unroll
    for (int b = 0; b < 7; ++b) {
      int col = j + b - 7;
      float p = pat[a * 7 + b];
      float v = (row >= 0 && col >= 0) ? x[row * IMG_W + col] : 0.f;
      acc  += fabsf(p - v);
      psum += fabsf(p);
    }
  }
  float raw = 1.f - acc  * (1.f / 49.f);
  float thr = 1.f - psum * (1.f / 49.f);
  F[idx] = (raw > thr) ? raw : 0.f;   // strict >, so the (0,0) element itself zeroes
}

// ---------------- Kernel 2: build Wmat[1536][224] ----------------
__global__ void wbuild_kernel(const float* __restrict__ vectors,
                              float* __restrict__ Wm) {
  int idx = blockIdx.x * blockDim.x + threadIdx.x;   // over IMG_H * KDIM
  int i   = idx / KDIM;
  int col = idx - i * KDIM;
  int k   = col / NROWS;
  int r   = col - k * NROWS;

  float v0 = vectors[2 * k + 0];
  float v1 = vectors[2 * k + 1];
  float vx = floorf(v0), vy = floorf(v1);
  int   sx = (int)vx,    sy = (int)vy;
  float px = vx - v0,    py = vy - v1;        // in (-1, 0]

  int   cx[4] = { sx,      sx,          sx + 1,        sx + 1 };
  int   cy[4] = { sy,      sy + 1,      sy,            sy + 1 };
  float cp[4] = { px * py, px * (1.f - py), (1.f - px) * py, (1.f - px) * (1.f - py) };

  float w = 0.f;
#pragma unroll
  for (int c = 0; c < 4; ++c) {
    int t = i - cy[c];
    t = t < 0 ? 0 : (t > 27 ? 27 : t);
    t = t - cx[c];
    t = t < 0 ? 0 : (t > 27 ? 27 : t);
    if (t == r) w += cp[c];
  }
  Wm[idx] = w * (1.f / 7.f);    // fold the /(K-1)
}

// ---------------- Kernel 3: WMMA f32 GEMM (1536x224x1536) + fused max ----------------
// One wave (32 lanes) owns one 16x16 output tile; 8 waves (256 threads) per block.
__global__ void gemm_max_kernel(const float* __restrict__ Wm,
                                const float* __restrict__ F,
                                float* __restrict__ tileMax) {
  int lane = threadIdx.x & 31;
  int wave = threadIdx.x >> 5;
  int tile = blockIdx.x * 8 + wave;
  int it = tile / NTILE;
  int jt = tile - it * NTILE;

  int mn    = lane & 15;          // M index for A frag, N index for B frag
  int khalf = (lane >> 4) * 2;    // lanes 0-15 -> K 0,1 ; lanes 16-31 -> K 2,3 (ISA 7.12.2)

  const float* Arow  = Wm + (it * 16 + mn) * KDIM + khalf;
  const float* Bbase = F  + jt * 16 + mn;

  v8f acc = {};
  for (int kk = 0; kk < KDIM; kk += 4) {
    v2f a, b;
    a.x = Arow[kk + 0];
    a.y = Arow[kk + 1];
    b.x = Bbase[(kk + khalf + 0) * IMG_W];
    b.y = Bbase[(kk + khalf + 1) * IMG_W];
    // D = A(16x4,f32) * B(4x16,f32) + C ; full f32 precision matrix path on gfx1250
    acc = __builtin_amdgcn_wmma_f32_16x16x4_f32(
        /*neg_a=*/false, a, /*neg_b=*/false, b,
        /*c_mod=*/(short)0, acc, /*reuse_a=*/false, /*reuse_b=*/false);
  }

  float mx = acc[0];
#pragma unroll
  for (int t = 1; t < 8; ++t) mx = fmaxf(mx, acc[t]);
#pragma unroll
  for (int off = 16; off > 0; off >>= 1)
    mx = fmaxf(mx, __shfl_xor(mx, off, 32));
  if (lane == 0) tileMax[tile] = mx;
}

// ---------------- Kernel 4: final max reduce ----------------
__global__ void reduce_max_kernel(const float* __restrict__ tileMax,
                                  float* __restrict__ out, int n) {
  __shared__ float red[256];
  float m = -3.402823466e+38f;
  for (int t = threadIdx.x; t < n; t += 256) m = fmaxf(m, tileMax[t]);
  red[threadIdx.x] = m;
  __syncthreads();
  for (int s = 128; s > 0; s >>= 1) {
    if ((int)threadIdx.x < s)
      red[threadIdx.x] = fmaxf(red[threadIdx.x], red[threadIdx.x + s]);
    __syncthreads();
  }
  if (threadIdx.x == 0) out[0] = red[0];
}

extern "C" void kernel_launch(void* const* d_in, const int* in_sizes, int n_in,
                              void* d_out, int out_size, void* d_ws, size_t ws_size,
                              hipStream_t stream) {
  const float* x        = (const float*)d_in[0];   // (1,1,1536,1536) f32
  const float* patterns = (const float*)d_in[1];   // (8,7,7) f32
  const float* vectors  = (const float*)d_in[2];   // (8,2) f32
  float* out = (float*)d_out;                      // scalar f32

  float* ws      = (float*)d_ws;
  float* F       = ws;                             // 8*28*1536   = 344064 floats
  float* Wm      = F  + KPAT * NROWS * IMG_W;      // 1536*224    = 344064 floats
  float* tileMax = Wm + IMG_H * KDIM;              // 96*96       = 9216 floats

  const int nA = KPAT * NROWS * IMG_W;             // 344064 -> 1344 blocks
  geneo1_slice_kernel<<<nA / 256, 256, 0, stream>>>(x, patterns, F);

  const int nW = IMG_H * KDIM;                     // 344064 -> 1344 blocks
  wbuild_kernel<<<nW / 256, 256, 0, stream>>>(vectors, Wm);

  const int ntiles = NTILE * NTILE;                // 9216 tiles, 8 waves/block
  gemm_max_kernel<<<ntiles / 8, 256, 0, stream>>>(Wm, F, tileMax);

  reduce_max_kernel<<<1, 256, 0, stream>>>(tileMax, out, ntiles);
}